// GATv2_28836410425871
// MI455X (gfx1250) — compile-verified
//
#include <hip/hip_runtime.h>
#include <math.h>

#define N_NODES 50000
#define N_EDGES 800000
#define ETOT    (N_EDGES + N_NODES)
#define DIM     128
#define LAYERS  3
#define SLOPE   0.2f

typedef __attribute__((ext_vector_type(2))) float v2f;
typedef __attribute__((ext_vector_type(8))) float v8f;

// ---------------------------------------------------------------------------
// GEMM via V_WMMA_F32_16X16X4_F32: Y = X(50000x128) @ W(128x128) + b
// One wave computes a full 16x128 row stripe: 8 persistent accumulators,
// A fragment loaded once per K-step and reused across 8 column-tile WMMAs
// (8x less A traffic; 8 independent WMMAs per K-step hide XDL latency).
// blockIdx.z selects (Wl,bl,Yl) vs (Wr,br,Yr).
// ---------------------------------------------------------------------------
__global__ __launch_bounds__(32)
void gatv2_gemm_wmma(const float* __restrict__ X,
                     const float* __restrict__ Wl, const float* __restrict__ bl,
                     const float* __restrict__ Wr, const float* __restrict__ br,
                     float* __restrict__ Yl, float* __restrict__ Yr) {
    const float* W = blockIdx.z ? Wr : Wl;
    const float* b = blockIdx.z ? br : bl;
    float*       Y = blockIdx.z ? Yr : Yl;

    const int lane = threadIdx.x & 31;
    const int row0 = blockIdx.x * 16;
    const int half = lane >> 4;          // 0: lanes 0-15, 1: lanes 16-31
    const int lid  = lane & 15;
    const int mrow = row0 + lid;         // A-fragment row

    v8f acc[8];
    #pragma unroll
    for (int t = 0; t < 8; ++t) acc[t] = (v8f){};

    for (int k = 0; k < DIM; k += 4) {
        const int ka = k + 2 * half;     // VGPR0 holds K=ka, VGPR1 holds K=ka+1
        v2f a;
        a.x = X[mrow * DIM + ka];
        a.y = X[mrow * DIM + ka + 1];
        #pragma unroll
        for (int t = 0; t < 8; ++t) {
            v2f bb;
            bb.x = W[ka       * DIM + t * 16 + lid];
            bb.y = W[(ka + 1) * DIM + t * 16 + lid];
            acc[t] = __builtin_amdgcn_wmma_f32_16x16x4_f32(
                         false, a, false, bb, (short)0, acc[t], false, false);
        }
    }

    #pragma unroll
    for (int t = 0; t < 8; ++t) {
        const float bias = b[t * 16 + lid];
        #pragma unroll
        for (int r = 0; r < 8; ++r) {
            const int row = row0 + r + 8 * half;  // C/D layout: VGPR r -> M=r / M=r+8
            Y[row * DIM + t * 16 + lid] = acc[t][r] + bias;
        }
    }
}

// ---------------------------------------------------------------------------
// Per-layer init: zero accumulation target, reset segment max/denominator.
// Encoded -inf for order-preserving uint max: ~bits(-inf) = 0x007FFFFF.
// ---------------------------------------------------------------------------
__global__ void gatv2_init_layer(float* __restrict__ hout,
                                 unsigned* __restrict__ menc,
                                 float* __restrict__ denom) {
    const int i = blockIdx.x * blockDim.x + threadIdx.x;
    if (i < N_NODES * DIM) hout[i] = 0.0f;
    if (i < N_NODES) { menc[i] = 0x007FFFFFu; denom[i] = 0.0f; }
}

// ---------------------------------------------------------------------------
// Edge scores: score[e] = att . leaky_relu(xl[src] + xr[dst]).
// One wave (32 lanes) per edge; float4 per lane = one coalesced 512B row.
// ---------------------------------------------------------------------------
__global__ void gatv2_edge_score(const float* __restrict__ xl,
                                 const float* __restrict__ xr,
                                 const int* __restrict__ src,
                                 const int* __restrict__ dst,
                                 const float* __restrict__ att,
                                 float* __restrict__ score) {
    const int e = blockIdx.x * (blockDim.x >> 5) + (threadIdx.x >> 5);
    if (e >= ETOT) return;
    const int lane = threadIdx.x & 31;

    int s, d;
    if (e < N_EDGES) { s = src[e]; d = dst[e]; } else { s = d = e - N_EDGES; }

    const float4 a = ((const float4*)(xl + (size_t)s * DIM))[lane];
    const float4 c = ((const float4*)(xr + (size_t)d * DIM))[lane];
    const float4 t = ((const float4*)att)[lane];

    float acc = 0.0f, v;
    v = a.x + c.x; acc += t.x * (v > 0.0f ? v : v * SLOPE);
    v = a.y + c.y; acc += t.y * (v > 0.0f ? v : v * SLOPE);
    v = a.z + c.z; acc += t.z * (v > 0.0f ? v : v * SLOPE);
    v = a.w + c.w; acc += t.w * (v > 0.0f ? v : v * SLOPE);

    for (int off = 16; off > 0; off >>= 1) acc += __shfl_down(acc, off, 32);
    if (lane == 0) score[e] = acc;
}

// ---------------------------------------------------------------------------
// Segment max over dst via order-preserving uint encoding + atomicMax.
// ---------------------------------------------------------------------------
__global__ void gatv2_seg_max(const float* __restrict__ score,
                              const int* __restrict__ dst,
                              unsigned* __restrict__ menc) {
    const int e = blockIdx.x * blockDim.x + threadIdx.x;
    if (e >= ETOT) return;
    const int d = (e < N_EDGES) ? dst[e] : e - N_EDGES;
    const unsigned u = __float_as_uint(score[e]);
    const unsigned enc = (u & 0x80000000u) ? ~u : (u | 0x80000000u);
    atomicMax(&menc[d], enc);
}

// ---------------------------------------------------------------------------
// p = exp(score - max[dst]); denom[dst] += p. p overwrites score in place.
// ---------------------------------------------------------------------------
__global__ void gatv2_softmax_num(float* __restrict__ score,
                                  const unsigned* __restrict__ menc,
                                  const int* __restrict__ dst,
                                  float* __restrict__ denom) {
    const int e = blockIdx.x * blockDim.x + threadIdx.x;
    if (e >= ETOT) return;
    const int d = (e < N_EDGES) ? dst[e] : e - N_EDGES;
    const unsigned u = menc[d];
    const float m = (u & 0x80000000u) ? __uint_as_float(u ^ 0x80000000u)
                                      : __uint_as_float(~u);
    const float p = expf(score[e] - m);
    score[e] = p;
    atomicAdd(&denom[d], p);
}

// ---------------------------------------------------------------------------
// Weighted scatter-aggregate: hout[dst] += xl[src] * (p / (denom[dst]+eps)).
// One wave per edge; 4 atomic f32 adds per lane on consecutive addresses.
// ---------------------------------------------------------------------------
__global__ void gatv2_aggregate(const float* __restrict__ xl,
                                const float* __restrict__ p,
                                const float* __restrict__ denom,
                                const int* __restrict__ src,
                                const int* __restrict__ dst,
                                float* __restrict__ hout) {
    const int e = blockIdx.x * (blockDim.x >> 5) + (threadIdx.x >> 5);
    if (e >= ETOT) return;
    const int lane = threadIdx.x & 31;

    int s, d;
    if (e < N_EDGES) { s = src[e]; d = dst[e]; } else { s = d = e - N_EDGES; }

    const float alpha = p[e] / (denom[d] + 1e-16f);
    const float4 v = ((const float4*)(xl + (size_t)s * DIM))[lane];
    float* o = hout + (size_t)d * DIM + lane * 4;
    atomicAdd(o + 0, v.x * alpha);
    atomicAdd(o + 1, v.y * alpha);
    atomicAdd(o + 2, v.z * alpha);
    atomicAdd(o + 3, v.w * alpha);
}

// ---------------------------------------------------------------------------
// In-place epilogue: h = relu(h + conv_bias)
// ---------------------------------------------------------------------------
__global__ void gatv2_relu_bias(float* __restrict__ h,
                                const float* __restrict__ bias) {
    const int i = blockIdx.x * blockDim.x + threadIdx.x;
    if (i >= N_NODES * DIM) return;
    const float v = h[i] + bias[i & (DIM - 1)];
    h[i] = v > 0.0f ? v : 0.0f;
}

// ---------------------------------------------------------------------------
// Row-wise log_softmax over DIM=128. One wave per node.
// ---------------------------------------------------------------------------
__global__ void gatv2_log_softmax(const float* __restrict__ h,
                                  float* __restrict__ out) {
    const int n = blockIdx.x * (blockDim.x >> 5) + (threadIdx.x >> 5);
    if (n >= N_NODES) return;
    const int lane = threadIdx.x & 31;

    const float4 v = ((const float4*)(h + (size_t)n * DIM))[lane];
    float mx = fmaxf(fmaxf(v.x, v.y), fmaxf(v.z, v.w));
    for (int off = 16; off > 0; off >>= 1) mx = fmaxf(mx, __shfl_xor(mx, off, 32));

    float s = expf(v.x - mx) + expf(v.y - mx) + expf(v.z - mx) + expf(v.w - mx);
    for (int off = 16; off > 0; off >>= 1) s += __shfl_xor(s, off, 32);

    const float lse = mx + logf(s);
    float4 o;
    o.x = v.x - lse; o.y = v.y - lse; o.z = v.z - lse; o.w = v.w - lse;
    ((float4*)(out + (size_t)n * DIM))[lane] = o;
}

// ---------------------------------------------------------------------------
extern "C" void kernel_launch(void* const* d_in, const int* in_sizes, int n_in,
                              void* d_out, int out_size, void* d_ws, size_t ws_size,
                              hipStream_t stream) {
    const float* x    = (const float*)d_in[0];
    const int*   ei   = (const int*)  d_in[1];   // (2, E)
    const float* Wl   = (const float*)d_in[2];   // (L, D, D)
    const float* bl   = (const float*)d_in[3];   // (L, D)
    const float* Wr   = (const float*)d_in[4];   // (L, D, D)
    const float* br   = (const float*)d_in[5];   // (L, D)
    const float* att  = (const float*)d_in[6];   // (L, 1, D)
    const float* bias = (const float*)d_in[7];   // (L, D)
    float* out        = (float*)d_out;

    const int* srcArr = ei;
    const int* dstArr = ei + N_EDGES;

    // Workspace carving
    const size_t ND = (size_t)N_NODES * DIM;
    float*    hA    = (float*)d_ws;
    float*    hB    = hA + ND;
    float*    xl    = hB + ND;
    float*    xr    = xl + ND;
    float*    score = xr + ND;                 // ETOT floats
    unsigned* menc  = (unsigned*)(score + ETOT);
    float*    denom = (float*)(menc + N_NODES);

    const dim3 gemmGrid(N_NODES / 16, 1, 2);         // 3125 x 1 x 2 row-stripe waves
    const int  edgeWaveBlocks = (ETOT + 7) / 8;      // 8 edges (waves) / 256-thr block
    const int  edgeThrBlocks  = (ETOT + 255) / 256;
    const int  ndBlocks       = (int)((ND + 255) / 256);
    const int  nodeWaveBlocks = (N_NODES + 7) / 8;

    const float* hin  = x;
    float*       hout = hA;

    for (int l = 0; l < LAYERS; ++l) {
        const float* Wl_l = Wl  + (size_t)l * DIM * DIM;
        const float* Wr_l = Wr  + (size_t)l * DIM * DIM;
        const float* bl_l = bl  + (size_t)l * DIM;
        const float* br_l = br  + (size_t)l * DIM;
        const float* at_l = att + (size_t)l * DIM;
        const float* bi_l = bias+ (size_t)l * DIM;

        gatv2_gemm_wmma<<<gemmGrid, 32, 0, stream>>>(hin, Wl_l, bl_l, Wr_l, br_l, xl, xr);
        gatv2_init_layer<<<ndBlocks, 256, 0, stream>>>(hout, menc, denom);
        gatv2_edge_score<<<edgeWaveBlocks, 256, 0, stream>>>(xl, xr, srcArr, dstArr, at_l, score);
        gatv2_seg_max<<<edgeThrBlocks, 256, 0, stream>>>(score, dstArr, menc);
        gatv2_softmax_num<<<edgeThrBlocks, 256, 0, stream>>>(score, menc, dstArr, denom);
        gatv2_aggregate<<<edgeWaveBlocks, 256, 0, stream>>>(xl, score, denom, srcArr, dstArr, hout);
        gatv2_relu_bias<<<ndBlocks, 256, 0, stream>>>(hout, bi_l);

        hin  = hout;
        hout = (hout == hA) ? hB : hA;
    }

    gatv2_log_softmax<<<nodeWaveBlocks, 256, 0, stream>>>(hin, out);
}